// DepthModule_16561393893775
// MI455X (gfx1250) — compile-verified
//
#include <hip/hip_runtime.h>
#include <math.h>

typedef __attribute__((ext_vector_type(2))) float v2f;
typedef __attribute__((ext_vector_type(8))) float v8f;

#define BSZ    4
#define NPTS   8192
#define SSTEPS 128
#define HIDN   64
#define CDIMN  128
#define NSEC   8

// single-instruction relu: v_med3_f32(x, 0, +inf) == max(x, 0) without the
// canonicalizing v_max_num(x,x) that fmaxf emits
__device__ __forceinline__ float relu1(float x) {
    return __builtin_amdgcn_fmed3f(x, 0.0f, __builtin_inff());
}

// ---------------------------------------------------------------------------
// Kernel 1: cc[b][h] = c[b,:] @ Wc[:,h] + b1[h]   (4 x 64 floats into d_ws)
// ---------------------------------------------------------------------------
__global__ __launch_bounds__(256) void cc_kernel(const float* __restrict__ c,
                                                 const float* __restrict__ Wc,
                                                 const float* __restrict__ b1,
                                                 float* __restrict__ cc) {
    int t = threadIdx.x;          // 0..255
    int b = t >> 6;
    int h = t & 63;
    float acc = b1[h];
    for (int k = 0; k < CDIMN; ++k)
        acc = fmaf(c[b * CDIMN + k], Wc[k * HIDN + h], acc);
    cc[b * HIDN + h] = acc;
}

// ---------------------------------------------------------------------------
// Kernel 2: one wave32 owns 32 rays (= two 16-ray WMMA tiles).
// Decoder eval: D(16x16) = A(16x4: x,y,z,1) * B(4x16: W1 rows + cc-bias row)
// via V_WMMA_F32_16X16X4_F32, 4 chunks to cover HID=64, then per-lane
// relu+fma with W2 and a butterfly multi-reduction so lane r holds val[ray r].
// ---------------------------------------------------------------------------
__global__ __launch_bounds__(256) void depth_kernel(
    const float* __restrict__ ray0, const float* __restrict__ rdir,
    const float* __restrict__ W1,   const float* __restrict__ W2,
    const float* __restrict__ b2p,  const float* __restrict__ cc,
    float* __restrict__ out)
{
    const int lane    = threadIdx.x & 31;
    const int wave    = (int)((blockIdx.x * blockDim.x + threadIdx.x) >> 5);
    const int rayBase = wave * 32;
    const int myRay   = rayBase + lane;
    const int b       = myRay / NPTS;
    const int l       = lane & 15;
    const bool hi     = lane >= 16;

    // --- B matrix (4x16 per chunk, 2 VGPRs): row0=W1.x, row1=W1.y,
    //     row2=W1.z, row3=cc(b)+b1 (bias via augmented 1 in A) ---
    const float* ccb = cc + b * HIDN;
    float B0[4], B1[4], w2l[4];
    #pragma unroll
    for (int ch = 0; ch < 4; ++ch) {
        int h = ch * 16 + l;
        B0[ch]  = hi ? W1[2 * HIDN + h] : W1[0 * HIDN + h];  // K=2 : K=0
        B1[ch]  = hi ? ccb[h]           : W1[1 * HIDN + h];  // K=3 : K=1
        w2l[ch] = W2[ch * 16 + l];
    }
    const float b2 = b2p[0];

    // --- per-lane A-build constants for both tiles ---
    // tile0 = rays rayBase+0..15, tile1 = rays rayBase+16..31
    int r0 = rayBase + l, r1 = rayBase + 16 + l;
    float o0x = ray0[r0*3+0], o0y = ray0[r0*3+1], o0z = ray0[r0*3+2];
    float e0x = rdir[r0*3+0], e0y = rdir[r0*3+1], e0z = rdir[r0*3+2];
    float o1x = ray0[r1*3+0], o1y = ray0[r1*3+1], o1z = ray0[r1*3+2];
    float e1x = rdir[r1*3+0], e1y = rdir[r1*3+1], e1z = rdir[r1*3+2];

    // A layout (16x4 f32): lanes<16 hold (K0=x, K1=y); lanes>=16 hold (K2=z, K3=1)
    float oA0t0 = hi ? o0z : o0x, vA0t0 = hi ? e0z : e0x;
    float oA1t0 = hi ? 1.0f : o0y, vA1t0 = hi ? 0.0f : e0y;
    float oA0t1 = hi ? o1z : o1x, vA0t1 = hi ? e1z : e1x;
    float oA1t1 = hi ? 1.0f : o1y, vA1t1 = hi ? 0.0f : e1y;

    // butterfly multi-reduction: 8 regs x 16 lanes -> lane (l&7) gets sum of
    // reg (l&7); halves untouched (xor masks 1,2,4,8 stay within a half)
    auto reduceTile = [&](const float (&t)[8]) -> float {
        float u0 = ((lane&1) ? t[1] : t[0]) + __shfl_xor((lane&1) ? t[0] : t[1], 1, 32);
        float u1 = ((lane&1) ? t[3] : t[2]) + __shfl_xor((lane&1) ? t[2] : t[3], 1, 32);
        float u2 = ((lane&1) ? t[5] : t[4]) + __shfl_xor((lane&1) ? t[4] : t[5], 1, 32);
        float u3 = ((lane&1) ? t[7] : t[6]) + __shfl_xor((lane&1) ? t[6] : t[7], 1, 32);
        float v0 = ((lane&2) ? u1 : u0) + __shfl_xor((lane&2) ? u0 : u1, 2, 32);
        float v1 = ((lane&2) ? u3 : u2) + __shfl_xor((lane&2) ? u2 : u3, 2, 32);
        float w  = ((lane&4) ? v1 : v0) + __shfl_xor((lane&4) ? v0 : v1, 4, 32);
        return w + __shfl_xor(w, 8, 32);
    };

    // decoder eval at per-tile depths; returns occupancy logit for lane's ray
    auto evalAt = [&](float dT0, float dT1) -> float {
        v2f A0, A1;
        A0.x = fmaf(dT0, vA0t0, oA0t0);  A0.y = fmaf(dT0, vA1t0, oA1t0);
        A1.x = fmaf(dT1, vA0t1, oA0t1);  A1.y = fmaf(dT1, vA1t1, oA1t1);
        float t0[8] = {0,0,0,0,0,0,0,0};
        float t1[8] = {0,0,0,0,0,0,0,0};
        #pragma unroll
        for (int ch = 0; ch < 4; ++ch) {
            v2f Bv; Bv.x = B0[ch]; Bv.y = B1[ch];
            v8f z = {};
            v8f C0 = __builtin_amdgcn_wmma_f32_16x16x4_f32(
                false, A0, false, Bv, (short)0, z, false, false);
            v8f C1 = __builtin_amdgcn_wmma_f32_16x16x4_f32(
                false, A1, false, Bv, (short)0, z, false, false);
            #pragma unroll
            for (int i = 0; i < 8; ++i) {
                t0[i] = fmaf(relu1(C0[i]), w2l[ch], t0[i]);
                t1[i] = fmaf(relu1(C1[i]), w2l[ch], t1[i]);
            }
        }
        float fA = reduceTile(t0);
        float fB = reduceTile(t1);
        float xA = __shfl_xor(fA, 16, 32);
        float xB = __shfl_xor(fB, 16, 32);
        // route so lane r ends with val[ray rayBase + r]
        float val = hi ? ((lane & 8) ? fB : xB)
                       : ((lane & 8) ? xA : fA);
        return val + b2;   // LOGIT_TAU == 0 for tau=0.5
    };

    // --- coarse ray marching: first - -> + sign change == argmin of cost ---
    const float dstep = 2.4f / (float)(SSTEPS - 1);
    float vprev = 0.0f;
    bool  found = false, mask0 = false;
    float f_low = -1.0f, f_high = 1.0f, d_low = 0.0f, d_high = 1.0f;
    for (int s = 0; s < SSTEPS; ++s) {
        float ds = (float)s * dstep;
        float v  = evalAt(ds, ds);
        if (s == 0) {
            mask0 = (v < 0.0f);
        } else {
            float sp = vprev * v;
            bool u = (sp < 0.0f) && !found;
            f_low  = u ? vprev : f_low;
            f_high = u ? v     : f_high;
            d_low  = u ? (float)(s - 1) * dstep : d_low;
            d_high = u ? ds    : d_high;
            found  = found || (sp < 0.0f);
        }
        vprev = v;
    }
    bool mask = found && (f_low < 0.0f) && mask0;
    d_low  = mask ? d_low  :  0.0f;   // reference placeholders
    d_high = mask ? d_high :  1.0f;
    f_low  = mask ? f_low  : -1.0f;
    f_high = mask ? f_high :  1.0f;

    // --- secant refinement (all lanes participate; EXEC stays full) ---
    float d_pred = fmaf(-f_low, (d_high - d_low) / (f_high - f_low), d_low);
    for (int it = 0; it < NSEC; ++it) {
        float dO = d_pred;
        float dX = __shfl_xor(dO, 16, 32);
        float f_mid = evalAt(hi ? dX : dO,   // tile0 wants d of ray (lane&15)
                             hi ? dO : dX);  // tile1 wants d of ray 16+(lane&15)
        bool indLow = (f_mid < 0.0f);
        d_low  = indLow ? d_pred : d_low;
        f_low  = indLow ? f_mid  : f_low;
        d_high = indLow ? d_high : d_pred;
        f_high = indLow ? f_high : f_mid;
        d_pred = fmaf(-f_low, (d_high - d_low) / (f_high - f_low), d_low);
    }

    float res = mask  ? d_pred : __builtin_inff();
    res       = mask0 ? res    : 0.0f;
    out[myRay] = res;
}

// ---------------------------------------------------------------------------
extern "C" void kernel_launch(void* const* d_in, const int* in_sizes, int n_in,
                              void* d_out, int out_size, void* d_ws, size_t ws_size,
                              hipStream_t stream) {
    const float* ray0 = (const float*)d_in[0];  // (B,N,3)
    const float* rdir = (const float*)d_in[1];  // (B,N,3)
    const float* c    = (const float*)d_in[2];  // (B,128)
    const float* W1   = (const float*)d_in[3];  // (3,64)
    const float* Wc   = (const float*)d_in[4];  // (128,64)
    const float* b1   = (const float*)d_in[5];  // (64,)
    const float* W2   = (const float*)d_in[6];  // (64,1)
    const float* b2   = (const float*)d_in[7];  // (1,)
    float* out = (float*)d_out;                 // (B,N) f32
    float* cc  = (float*)d_ws;                  // 4*64 floats scratch

    cc_kernel<<<1, 256, 0, stream>>>(c, Wc, b1, cc);

    const int totalRays = BSZ * NPTS;           // 32768, 32 rays per wave
    depth_kernel<<<totalRays / 256, 256, 0, stream>>>(
        ray0, rdir, W1, W2, b2, cc, out);
}